// GCNModel_29970281791971
// MI455X (gfx1250) — compile-verified
//
#include <hip/hip_runtime.h>
#include <hip/hip_bf16.h>
#include <stdint.h>

typedef __attribute__((ext_vector_type(2))) float v2f;
typedef __attribute__((ext_vector_type(8))) float v8f;

// ---------------- utility kernels ----------------

__global__ void fill_kernel(float* __restrict__ p, float v, int n) {
  int i = blockIdx.x * blockDim.x + threadIdx.x;
  if (i < n) p[i] = v;
}

__global__ void deg_accum_kernel(const int* __restrict__ dst, float* __restrict__ deg, int E) {
  int e = blockIdx.x * blockDim.x + threadIdx.x;
  if (e < E) atomicAdd(&deg[dst[e]], 1.0f);
}

__global__ void rsqrt_kernel(float* __restrict__ p, int n) {
  int i = blockIdx.x * blockDim.x + threadIdx.x;
  if (i < n) p[i] = rsqrtf(p[i]);
}

// W[k][n] (128x128 row-major) -> Wt[n][k] so both K-values of a B fragment are contiguous
__global__ void transpose128_kernel(const float* __restrict__ W, float* __restrict__ Wt) {
  int t = blockIdx.x * blockDim.x + threadIdx.x;   // 128*128
  if (t >= 128 * 128) return;
  int k = t >> 7, n = t & 127;
  Wt[n * 128 + k] = W[k * 128 + n];
}

// ---------------- WMMA GEMM: H[M,128] = (relu?)A[M,128] @ W[128,128] ----------------
// Block = 32 output rows, 8 waves; wave w owns cols [16w,16w+16) for BOTH 16-row
// half-tiles -> two independent accumulator chains sharing one B fragment.
// B read from transposed W (Wt[n][k]) as a single global_load_b64 per k-step.

template<bool RELU_IN>
__global__ __launch_bounds__(256) void gemm128_wmma(const float* __restrict__ A,
                                                    const float* __restrict__ Wt,
                                                    float* __restrict__ H) {
  __shared__ float As[32][129];   // padded: conflict-free column reads
  const int tid  = threadIdx.x;
  const int row0 = blockIdx.x * 32;

  // cooperative A-tile load (32x128), fuse input ReLU
  for (int i = tid; i < 32 * 128; i += 256) {
    int r = i >> 7, c = i & 127;
    float v = A[(size_t)(row0 + r) * 128 + c];
    if (RELU_IN) v = fmaxf(v, 0.0f);
    As[r][c] = v;
  }
  __syncthreads();

  const int lane = tid & 31;
  const int wave = tid >> 5;
  const int col0 = wave * 16;
  const int m    = lane & 15;        // row (A) / col (B,D) within tile
  const int kh   = (lane >> 4) * 2;  // lanes 0-15 -> K 0,1 ; lanes 16-31 -> K 2,3

  const float* wrow = Wt + (size_t)(col0 + m) * 128 + kh;  // contiguous in K

  v8f acc0 = {};   // rows row0 .. row0+15
  v8f acc1 = {};   // rows row0+16 .. row0+31
  #pragma unroll
  for (int kk = 0; kk < 128; kk += 4) {
    v2f a0, a1;
    a0.x = As[m][kk + kh + 0];
    a0.y = As[m][kk + kh + 1];
    a1.x = As[16 + m][kk + kh + 0];
    a1.y = As[16 + m][kk + kh + 1];
    v2f b = *(const v2f*)(wrow + kk);          // global_load_b64, L2-resident
    acc0 = __builtin_amdgcn_wmma_f32_16x16x4_f32(false, a0, false, b,
                                                 (short)0, acc0, false, false);
    acc1 = __builtin_amdgcn_wmma_f32_16x16x4_f32(false, a1, false, b,
                                                 (short)0, acc1, false, false);
  }

  // D layout: VGPR r -> M=r (lanes 0-15) or M=8+r (lanes 16-31), N = lane&15
  const int rofs = (lane >> 4) * 8;
  #pragma unroll
  for (int r = 0; r < 8; ++r) {
    H[(size_t)(row0 + rofs + r) * 128 + col0 + m]      = acc0[r];
    H[(size_t)(row0 + 16 + rofs + r) * 128 + col0 + m] = acc1[r];
  }
}

// ---------------- self-loop + bias: agg = h * dinv^2 + b ----------------

__global__ void self_bias_kernel(const float* __restrict__ h, const float* __restrict__ dinv,
                                 const float* __restrict__ b, float* __restrict__ agg, int N) {
  long long t = (long long)blockIdx.x * blockDim.x + threadIdx.x;
  int i = (int)(t >> 7);
  int c = (int)(t & 127);
  if (i >= N) return;
  float di = dinv[i];
  agg[t] = h[t] * di * di + b[c];
}

// ---------------- edge scatter: agg[dst] += h[src] * dinv[src]*dinv[dst] ----------------
// 32 threads per edge, float4 (16B) gather per thread, 4 float atomics.

__global__ void edge_scatter_kernel(const int* __restrict__ src, const int* __restrict__ dst,
                                    const float* __restrict__ dinv, const float* __restrict__ h,
                                    float* __restrict__ agg, int E) {
  long long t = (long long)blockIdx.x * blockDim.x + threadIdx.x;
  int e  = (int)(t >> 5);
  int c4 = (int)(t & 31);
  if (e >= E) return;
  int s = src[e], d = dst[e];
  float nrm = dinv[s] * dinv[d];
  float4 v = ((const float4*)(h + (size_t)s * 128))[c4];
  float* ap = agg + (size_t)d * 128 + (size_t)c4 * 4;
  atomicAdd(ap + 0, v.x * nrm);
  atomicAdd(ap + 1, v.y * nrm);
  atomicAdd(ap + 2, v.z * nrm);
  atomicAdd(ap + 3, v.w * nrm);
}

// ---------------- global mean pool (ReLU fused) ----------------

__global__ void pool_accum_kernel(const float* __restrict__ agg, const int* __restrict__ batch,
                                  float* __restrict__ sums, float* __restrict__ counts, int N) {
  long long t = (long long)blockIdx.x * blockDim.x + threadIdx.x;
  int i = (int)(t >> 7);
  int c = (int)(t & 127);
  if (i >= N) return;
  float v = fmaxf(agg[t], 0.0f);
  int g = batch[i];
  atomicAdd(&sums[(size_t)g * 128 + c], v);
  if (c == 0) atomicAdd(&counts[g], 1.0f);
}

__global__ void pool_final_kernel(const float* __restrict__ sums, const float* __restrict__ counts,
                                  float* __restrict__ ge, int total) {
  int t = blockIdx.x * blockDim.x + threadIdx.x;
  if (t >= total) return;
  int g = t >> 7;
  ge[t] = sums[t] / fmaxf(counts[g], 1.0f);
}

// ---------------- FC: pred[128,64] = ge[128,128] @ Wfc[128,64] + bfc ----------------

__global__ void fc_kernel(const float* __restrict__ ge, const float* __restrict__ Wfc,
                          const float* __restrict__ bfc, float* __restrict__ pred) {
  int t = blockIdx.x * blockDim.x + threadIdx.x;  // 128*64
  if (t >= 128 * 64) return;
  int g = t >> 6, o = t & 63;
  float acc = bfc[o];
  #pragma unroll 8
  for (int k = 0; k < 128; ++k) acc += ge[g * 128 + k] * Wfc[k * 64 + o];
  pred[t] = acc;
}

// ---------------- launch ----------------

static inline unsigned cdivll(long long a, long long b) { return (unsigned)((a + b - 1) / b); }

extern "C" void kernel_launch(void* const* d_in, const int* in_sizes, int n_in,
                              void* d_out, int out_size, void* d_ws, size_t ws_size,
                              hipStream_t stream) {
  const float* x    = (const float*)d_in[0];
  const int*   ei   = (const int*)d_in[1];
  const int*   batch= (const int*)d_in[2];
  const float* W1   = (const float*)d_in[3];  const float* b1 = (const float*)d_in[4];
  const float* W2   = (const float*)d_in[5];  const float* b2 = (const float*)d_in[6];
  const float* W3   = (const float*)d_in[7];  const float* b3 = (const float*)d_in[8];
  const float* W4   = (const float*)d_in[9];  const float* b4 = (const float*)d_in[10];
  const float* Wfc  = (const float*)d_in[11]; const float* bfc= (const float*)d_in[12];

  const int N = in_sizes[2];          // 100000 nodes
  const int E = in_sizes[1] / 2;      // 1600000 edges
  const int* src = ei;
  const int* dst = ei + E;

  // workspace carve-up (floats)
  float* ws     = (float*)d_ws;
  float* dinv   = ws;                                   // N  (deg, then dinv in place)
  float* h      = dinv + ((N + 3) & ~3);                // N*128
  float* aggA   = h    + (size_t)N * 128;               // N*128
  float* aggB   = aggA + (size_t)N * 128;               // N*128
  float* sums   = aggB + (size_t)N * 128;               // 128*128
  float* counts = sums + 128 * 128;                     // 128
  float* Wt     = counts + 128;                         // 128*128 (transposed weights)

  float* ge   = (float*)d_out;          // [128,128]
  float* pred = ge + 128 * 128;         // [128,64]

  const int T = 256;
  const long long NC = (long long)N * 128;
  const unsigned gW = cdivll(128 * 128, T);

  // degrees with self loops -> dinv
  fill_kernel<<<cdivll(N, T), T, 0, stream>>>(dinv, 1.0f, N);
  deg_accum_kernel<<<cdivll(E, T), T, 0, stream>>>(dst, dinv, E);
  rsqrt_kernel<<<cdivll(N, T), T, 0, stream>>>(dinv, N);

  // layer 1: x -> aggA
  transpose128_kernel<<<gW, T, 0, stream>>>(W1, Wt);
  gemm128_wmma<false><<<N / 32, T, 0, stream>>>(x, Wt, h);
  self_bias_kernel<<<cdivll(NC, T), T, 0, stream>>>(h, dinv, b1, aggA, N);
  edge_scatter_kernel<<<cdivll((long long)E * 32, T), T, 0, stream>>>(src, dst, dinv, h, aggA, E);

  // layer 2: relu(aggA) -> aggB
  transpose128_kernel<<<gW, T, 0, stream>>>(W2, Wt);
  gemm128_wmma<true><<<N / 32, T, 0, stream>>>(aggA, Wt, h);
  self_bias_kernel<<<cdivll(NC, T), T, 0, stream>>>(h, dinv, b2, aggB, N);
  edge_scatter_kernel<<<cdivll((long long)E * 32, T), T, 0, stream>>>(src, dst, dinv, h, aggB, E);

  // layer 3: relu(aggB) -> aggA
  transpose128_kernel<<<gW, T, 0, stream>>>(W3, Wt);
  gemm128_wmma<true><<<N / 32, T, 0, stream>>>(aggB, Wt, h);
  self_bias_kernel<<<cdivll(NC, T), T, 0, stream>>>(h, dinv, b3, aggA, N);
  edge_scatter_kernel<<<cdivll((long long)E * 32, T), T, 0, stream>>>(src, dst, dinv, h, aggA, E);

  // layer 4: relu(aggA) -> aggB
  transpose128_kernel<<<gW, T, 0, stream>>>(W4, Wt);
  gemm128_wmma<true><<<N / 32, T, 0, stream>>>(aggA, Wt, h);
  self_bias_kernel<<<cdivll(NC, T), T, 0, stream>>>(h, dinv, b4, aggB, N);
  edge_scatter_kernel<<<cdivll((long long)E * 32, T), T, 0, stream>>>(src, dst, dinv, h, aggB, E);

  // mean pool (ReLU fused) + FC
  fill_kernel<<<cdivll(128 * 128 + 128, T), T, 0, stream>>>(sums, 0.0f, 128 * 128 + 128);
  pool_accum_kernel<<<cdivll(NC, T), T, 0, stream>>>(aggB, batch, sums, counts, N);
  pool_final_kernel<<<cdivll(128 * 128, T), T, 0, stream>>>(sums, counts, ge, 128 * 128);
  fc_kernel<<<cdivll(128 * 64, T), T, 0, stream>>>(ge, Wfc, bfc, pred);
}